// YOLOV1LossV2_59124519797021
// MI455X (gfx1250) — compile-verified
//
#include <hip/hip_runtime.h>

// YOLOv1 loss for MI455X (gfx1250, wave32). Bandwidth-bound: ~193 MB in,
// 4 B out -> ideal ~8.3 us at 23.3 TB/s. Data path: double-buffered LDS
// tiles staged with GLOBAL_LOAD_ASYNC_TO_LDS_B128 (coalesced 512 B per wave
// instruction, ASYNCcnt-tracked), per-cell reads from LDS (2-way conflicts
// only). Deterministic two-stage reduction; final 32-lane sum via two
// chained v_wmma_f32_16x16x4_f32 (all-ones B => layout-agnostic, exact).

typedef __attribute__((ext_vector_type(2))) float v2f;
typedef __attribute__((ext_vector_type(8))) float v8f;
typedef int v4i __attribute__((vector_size(16)));            // int4 (vector_size)
typedef __attribute__((address_space(1))) v4i* gv4i_ptr;     // global int4*
typedef __attribute__((address_space(3))) v4i* lv4i_ptr;     // LDS int4*

#if __has_builtin(__builtin_amdgcn_global_load_async_to_lds_b128) && \
    __has_builtin(__builtin_amdgcn_s_wait_asynccnt)
#define HAVE_ASYNC_LDS 1
#else
#define HAVE_ASYNC_LDS 0
#endif

namespace {
constexpr int       kN          = 30;                 // 5*B + C
constexpr long long kBatch      = 16384;
constexpr long long kCells      = kBatch * 49;        // 802816
constexpr int       kThreads    = 256;
constexpr int       kBlocks     = 448;
constexpr int       kTilesPerBlk= 7;                  // 448*7*256 == kCells
constexpr int       kTileFloats = kThreads * kN;      // 7680 floats / tensor
constexpr int       kTileVec4   = kTileFloats / 4;    // 1920 float4
}

// Stage one 30720-byte tile (gsrc, 16B-aligned) into LDS buffer l.
__device__ __forceinline__ void stage_tile(const float* __restrict__ g,
                                           float* l, int tid) {
#if HAVE_ASYNC_LDS
#pragma unroll
  for (int k = 0; k < 8; ++k) {
    int i = tid + k * kThreads;
    if (i < kTileVec4) {                 // per-lane EXEC masking is fine
      __builtin_amdgcn_global_load_async_to_lds_b128(
          (gv4i_ptr)(g + 4 * i),         // global src, AS(1) int4*
          (lv4i_ptr)(l + 4 * i),         // LDS dst,    AS(3) int4*
          /*imm offset*/ 0, /*cpol*/ 0);
    }
  }
#else
#pragma unroll
  for (int k = 0; k < 8; ++k) {
    int i = tid + k * kThreads;
    if (i < kTileVec4) {
      float4 v = ((const float4*)g)[i];
      ((float4*)l)[i] = v;
    }
  }
#endif
}

__device__ __forceinline__ void stage_wait() {
#if HAVE_ASYNC_LDS
  __builtin_amdgcn_s_wait_asynccnt(0);
#endif
  __syncthreads();
}

// Per-cell loss; sp/st point at this cell's 30 floats in LDS (8B-aligned).
__device__ __forceinline__ float cell_loss(const float* sp, const float* st) {
  float P[kN], T[kN];
#pragma unroll
  for (int k = 0; k < kN / 2; ++k) {     // 15x ds_load_b64 each
    v2f a = ((const v2f*)sp)[k]; P[2*k] = a.x; P[2*k+1] = a.y;
    v2f b = ((const v2f*)st)[k]; T[2*k] = b.x; T[2*k+1] = b.y;
  }

  // Target box 0, denormalized: xy/S center, box = [xy - wh/2, xy + wh/2].
  float tcx = T[0] / 7.0f, tcy = T[1] / 7.0f;
  float tlx = tcx - 0.5f * T[2], tly = tcy - 0.5f * T[3];
  float trx = tcx + 0.5f * T[2], tr_y = tcy + 0.5f * T[3];
  float ta  = T[2] * T[3];

  float iou0 = 0.f, iou1 = 0.f;
#pragma unroll
  for (int b = 0; b < 2; ++b) {
    const int o = 5 * b;
    float pcx = P[o] / 7.0f, pcy = P[o+1] / 7.0f;
    float plx = pcx - 0.5f * P[o+2], ply = pcy - 0.5f * P[o+3];
    float prx = pcx + 0.5f * P[o+2], pry = pcy + 0.5f * P[o+3];
    float lx = fmaxf(plx, tlx), ly = fmaxf(ply, tly);
    float rx = fminf(prx, trx), ry = fminf(pry, tr_y);
    float w = fmaxf(rx - lx, 0.f), h = fmaxf(ry - ly, 0.f);
    float inter = w * h;
    float uni = P[o+2] * P[o+3] + ta - inter;
    float iou = (uni > 0.f) ? (inter / uni) : 0.f;
    if (b == 0) iou0 = iou; else iou1 = iou;
  }

  const int   j    = (iou1 > iou0) ? 1 : 0;   // jnp.argmax: ties -> box 0
  const float miou = fmaxf(iou0, iou1);

  float loss;
  if (T[4] > 0.f) {                           // responsible cell (m == 1)
    const int o = 5 * j;
    float dx = P[o]   - T[o];
    float dy = P[o+1] - T[o+1];
    float sw = sqrtf(P[o+2]) - sqrtf(T[o+2]);
    float sh = sqrtf(P[o+3]) - sqrtf(T[o+3]);
    float db = P[o+4] - miou;
    float cls = 0.f;
#pragma unroll
    for (int k = 10; k < kN; ++k) { float d = P[k] - T[k]; cls += d * d; }
    loss = 5.0f * (dx*dx + dy*dy + sw*sw + sh*sh) + db*db + cls;
  } else {                                    // no-object cell (nm == 1)
    float d0 = P[4] - T[4];
    float d1 = P[9] - T[9];
    loss = 0.5f * (d0*d0 + d1*d1);
  }
  return loss;
}

__global__ void __launch_bounds__(kThreads)
yolo_main(const float* __restrict__ pred, const float* __restrict__ tgt,
          float* __restrict__ partial) {
  __shared__ __align__(16) float sP[2][kTileFloats];   // 2 x 30720 B
  __shared__ __align__(16) float sT[2][kTileFloats];   // 2 x 30720 B
  __shared__ float red[kThreads];

  const int tid = threadIdx.x;
  const long long tile0 = (long long)blockIdx.x * kTilesPerBlk;

  // Prologue: stage tile 0 into buffer 0.
  stage_tile(pred + tile0 * kTileFloats, sP[0], tid);
  stage_tile(tgt  + tile0 * kTileFloats, sT[0], tid);
  stage_wait();

  float acc = 0.f;
  int buf = 0;
#pragma unroll 1
  for (int it = 0; it < kTilesPerBlk; ++it) {
    if (it + 1 < kTilesPerBlk) {               // async-prefetch next tile
      const long long nt = tile0 + it + 1;
      stage_tile(pred + nt * kTileFloats, sP[buf ^ 1], tid);
      stage_tile(tgt  + nt * kTileFloats, sT[buf ^ 1], tid);
    }
    acc += cell_loss(&sP[buf][tid * kN], &sT[buf][tid * kN]);
    stage_wait();                              // next buffer ready + visible
    buf ^= 1;
  }

  red[tid] = acc;
  __syncthreads();
#pragma unroll
  for (int s = kThreads / 2; s > 0; s >>= 1) { // fixed-order tree
    if (tid < s) red[tid] += red[tid + s];
    __syncthreads();
  }
  if (tid == 0) partial[blockIdx.x] = red[0];
}

// Single-wave finisher: strided partial sums, then cross-lane sum via two
// chained V_WMMA_F32_16X16X4_F32 (all-ones B). EXEC is all ones (whole wave).
__global__ void __launch_bounds__(32)
yolo_finish(const float* __restrict__ partial, float* __restrict__ out) {
  const int lane = threadIdx.x;               // 0..31, one wave32
  float v = 0.f;
  for (int i = lane; i < kBlocks; i += 32) v += partial[i];  // fixed order

  float total;
#if __has_builtin(__builtin_amdgcn_wmma_f32_16x16x4_f32)
  v2f a;   a.x = v;    a.y = v;     // A row m = [v_m, v_m, v_{m+16}, v_{m+16}]
  v2f one; one.x = 1.f; one.y = 1.f;
  v8f c = {};
  // D[m][n] = 2*(v_m + v_{m+16})
  v8f d = __builtin_amdgcn_wmma_f32_16x16x4_f32(false, a, false, one,
                                                (short)0, c, false, false);
  v2f a2;
  a2.x = d[0] + d[1] + d[2] + d[3];
  a2.y = d[4] + d[5] + d[6] + d[7];
  // D2[m][n] = sum_r D[r][m] = 2 * (sum over all 32 lanes of v)
  v8f d2 = __builtin_amdgcn_wmma_f32_16x16x4_f32(false, a2, false, one,
                                                 (short)0, c, false, false);
  total = 0.5f * d2[0];
#else
  total = v;
  for (int off = 16; off > 0; off >>= 1) total += __shfl_xor(total, off, 32);
#endif

  if (lane == 0) out[0] = total * (1.0f / (float)kBatch);
}

extern "C" void kernel_launch(void* const* d_in, const int* in_sizes, int n_in,
                              void* d_out, int out_size, void* d_ws, size_t ws_size,
                              hipStream_t stream) {
  const float* pred = (const float*)d_in[0];   // (16384,7,7,30) f32
  const float* tgt  = (const float*)d_in[1];   // (16384,7,7,30) f32
  float* partial = (float*)d_ws;               // 448 floats of scratch
  yolo_main<<<kBlocks, kThreads, 0, stream>>>(pred, tgt, partial);
  yolo_finish<<<1, 32, 0, stream>>>(partial, (float*)d_out);
  (void)in_sizes; (void)n_in; (void)out_size; (void)ws_size;
}